// RBWModel_35270271435277
// MI455X (gfx1250) — compile-verified
//
#include <hip/hip_runtime.h>
#include <math.h>

#define HID    512
#define FFD    1024
#define BB     32
#define TT     1024
#define VOCAB  32000
#define NCAND  1021          // T - 3
#define MSLOTS 64
#define THRESH 0.8f

typedef __attribute__((ext_vector_type(2))) float v2f;
typedef __attribute__((ext_vector_type(8))) float v8f;

// ---- workspace layout (in floats) ----
static const size_t OFF_HIDDEN = 0;                                   // 32*1024*512
static const size_t OFF_SCORES = OFF_HIDDEN + (size_t)BB*TT*HID;      // 32*1021
static const size_t OFF_MEM    = OFF_SCORES + (size_t)BB*NCAND;       // 32*64*512
static const size_t OFF_MASK   = OFF_MEM + (size_t)BB*MSLOTS*HID;     // 32*64
static const size_t OFF_CTX    = OFF_MASK + (size_t)BB*MSLOTS;        // 32*512

// =====================================================================
// K1: fused embed-gather (async-to-LDS) + FFN + residual + LayerNorm
// one block = 32 tokens; 8 waves; WMMA f32 16x16x4 with 2x M-blocking
// (each B fragment feeds two WMMAs -> half the L2 weight traffic)
// =====================================================================
__global__ void __launch_bounds__(256)
ffn_ln_kernel(const int* __restrict__ seq, const float* __restrict__ embed,
              const float* __restrict__ w1, const float* __restrict__ b1,
              const float* __restrict__ w2, const float* __restrict__ b2,
              const float* __restrict__ ln_g, const float* __restrict__ ln_b,
              float* __restrict__ hidden)
{
    extern __shared__ float smem[];
    float* ldsA  = smem;               // 32 x 512  (gathered embeddings / residual)
    float* ldsFF = smem + 32*HID;      // 32 x 1024 (ReLU'd ffn intermediate)
    float* ldsX  = ldsFF + 32*FFD;     // 32 x 512  (pre-LN activations)

    const int tid  = threadIdx.x;
    const int wave = tid >> 5;
    const int lane = tid & 31;
    const int n    = lane & 15;        // M index for A-lanes, N index for B-lanes
    const int kh   = lane >> 4;        // K-half select (per ISA f32 WMMA layout)
    const size_t m0 = (size_t)blockIdx.x * 32;

    // stage the 32 token ids (borrow ldsX before it is written)
    int* s_tok = (int*)ldsX;
    if (tid < 32) s_tok[tid] = seq[m0 + tid];
    __syncthreads();

    // ---- async gather: 32 embedding rows -> ldsA via ASYNCcnt path ----
    {
        const unsigned ldsA_base = (unsigned)(unsigned long long)(void*)ldsA;
        for (int j = tid; j < 32 * (HID/4); j += 256) {     // 4096 x 16B chunks
            const int row = j >> 7;                          // j / 128
            const int c   = j & 127;
            const unsigned long long ga =
                (unsigned long long)(embed + (size_t)s_tok[row] * HID + c*4);
            const unsigned loff = ldsA_base + (unsigned)((row*HID + c*4) * 4);
            asm volatile("global_load_async_to_lds_b128 %0, %1, off"
                         :: "v"(loff), "v"(ga) : "memory");
        }
        asm volatile("s_wait_asynccnt 0x0" ::: "memory");
    }
    __syncthreads();

    // ---- GEMM1: ff = relu(A(32x512) * w1^T + b1) -> ldsFF ----
    for (int jt = wave; jt < FFD/16; jt += 8) {
        const int n0 = jt * 16;
        const float* Bp  = w1 + (size_t)(n0 + n) * HID + 2*kh;
        const float* Ap0 = ldsA + (size_t)n * HID + 2*kh;
        const float* Ap1 = ldsA + (size_t)(16 + n) * HID + 2*kh;
        if (jt + 8 < FFD/16)  // prefetch next weight tile (global_prefetch_b8)
            __builtin_prefetch(w1 + (size_t)(n0 + 128 + n) * HID, 0, 1);
        v8f acc0 = {0.f,0.f,0.f,0.f,0.f,0.f,0.f,0.f};
        v8f acc1 = {0.f,0.f,0.f,0.f,0.f,0.f,0.f,0.f};
        for (int k0 = 0; k0 < HID; k0 += 4) {
            v2f b  = *(const v2f*)(Bp  + k0);
            v2f a0 = *(const v2f*)(Ap0 + k0);
            v2f a1 = *(const v2f*)(Ap1 + k0);
            acc0 = __builtin_amdgcn_wmma_f32_16x16x4_f32(false, a0, false, b,
                                                         (short)0, acc0, false, false);
            acc1 = __builtin_amdgcn_wmma_f32_16x16x4_f32(false, a1, false, b,
                                                         (short)0, acc1, false, false);
        }
        const float bias = b1[n0 + n];
        #pragma unroll
        for (int r = 0; r < 8; ++r) {
            float v0 = acc0[r] + bias;
            float v1 = acc1[r] + bias;
            ldsFF[(r + 8*kh) * FFD + (n0 + n)]        = v0 > 0.f ? v0 : 0.f;
            ldsFF[(16 + r + 8*kh) * FFD + (n0 + n)]   = v1 > 0.f ? v1 : 0.f;
        }
    }
    __syncthreads();

    // ---- GEMM2: x = ff(32x1024) * w2^T + b2 + residual -> ldsX ----
    for (int jt = wave; jt < HID/16; jt += 8) {
        const int h0 = jt * 16;
        const float* Bp  = w2 + (size_t)(h0 + n) * FFD + 2*kh;
        const float* Ap0 = ldsFF + (size_t)n * FFD + 2*kh;
        const float* Ap1 = ldsFF + (size_t)(16 + n) * FFD + 2*kh;
        if (jt + 8 < HID/16)
            __builtin_prefetch(w2 + (size_t)(h0 + 128 + n) * FFD, 0, 1);
        v8f acc0 = {0.f,0.f,0.f,0.f,0.f,0.f,0.f,0.f};
        v8f acc1 = {0.f,0.f,0.f,0.f,0.f,0.f,0.f,0.f};
        for (int k0 = 0; k0 < FFD; k0 += 4) {
            v2f b  = *(const v2f*)(Bp  + k0);
            v2f a0 = *(const v2f*)(Ap0 + k0);
            v2f a1 = *(const v2f*)(Ap1 + k0);
            acc0 = __builtin_amdgcn_wmma_f32_16x16x4_f32(false, a0, false, b,
                                                         (short)0, acc0, false, false);
            acc1 = __builtin_amdgcn_wmma_f32_16x16x4_f32(false, a1, false, b,
                                                         (short)0, acc1, false, false);
        }
        const float bias = b2[h0 + n];
        #pragma unroll
        for (int r = 0; r < 8; ++r) {
            const int r0 = r + 8*kh;
            const int r1 = 16 + r + 8*kh;
            ldsX[r0 * HID + (h0 + n)] = acc0[r] + bias + ldsA[r0 * HID + (h0 + n)];
            ldsX[r1 * HID + (h0 + n)] = acc1[r] + bias + ldsA[r1 * HID + (h0 + n)];
        }
    }
    __syncthreads();

    // ---- LayerNorm: each wave handles 4 rows ----
    for (int rr = 0; rr < 4; ++rr) {
        const int row = wave * 4 + rr;
        const float* xr = ldsX + row * HID;
        float s = 0.f, s2 = 0.f;
        #pragma unroll
        for (int i = 0; i < 16; ++i) {
            float v = xr[lane*16 + i];
            s += v; s2 += v*v;
        }
        for (int o = 16; o > 0; o >>= 1) {
            s  += __shfl_xor(s,  o, 32);
            s2 += __shfl_xor(s2, o, 32);
        }
        const float mu   = s  * (1.f/HID);
        const float var  = s2 * (1.f/HID) - mu*mu;
        const float rinv = rsqrtf(var + 1e-5f);
        float* outr = hidden + (m0 + row) * HID;
        #pragma unroll
        for (int i = 0; i < 16; ++i) {
            const int idx = lane*16 + i;
            outr[idx] = (xr[idx] - mu) * rinv * ln_g[idx] + ln_b[idx];
        }
    }
}

// =====================================================================
// K2: gate scores  scores[b,t] = hidden[b,t,:] . wg_w + wg_b
// =====================================================================
__global__ void __launch_bounds__(256)
scores_kernel(const float* __restrict__ hidden, const float* __restrict__ wg_w,
              const float* __restrict__ wg_b, float* __restrict__ scores)
{
    const int wave = threadIdx.x >> 5, lane = threadIdx.x & 31;
    const int gid = blockIdx.x * 8 + wave;
    if (gid >= BB * NCAND) return;
    const int b = gid / NCAND, t = gid % NCAND;
    const float* hrow = hidden + ((size_t)b * TT + t) * HID;
    float s = 0.f;
    for (int k = lane; k < HID; k += 32) s += hrow[k] * wg_w[k];
    for (int o = 16; o > 0; o >>= 1) s += __shfl_xor(s, o, 32);
    if (lane == 0) scores[(size_t)b * NCAND + t] = s + wg_b[0];
}

// =====================================================================
// K3: per-row greedy score-ranked selection with cosine dedup
// =====================================================================
__global__ void __launch_bounds__(256)
select_kernel(const float* __restrict__ hidden, const float* __restrict__ scores,
              float* __restrict__ mem, float* __restrict__ mask)
{
    extern __shared__ float smem[];
    float* s_key  = smem;                  // 1024
    int*   s_ind  = (int*)(smem + 1024);   // 1024
    float* s_cvec = smem + 2048;           // 512
    float* s_red  = smem + 2560;           // 256
    float* s_sims = smem + 2816;           // 64
    float* s_bc   = smem + 2880;           // 4
    float* s_norm = smem + 2944;           // 64*512 (normalized selected)

    const int tid = threadIdx.x;
    const int b   = blockIdx.x;

    for (int i = tid; i < MSLOTS*HID; i += 256) {
        mem[(size_t)b*MSLOTS*HID + i] = 0.f;
        s_norm[i] = 0.f;
    }
    if (tid < MSLOTS) mask[b*MSLOTS + tid] = 0.f;

    // keys = -score (ascending sort == argsort(-scores)); pad with +inf
    for (int i = tid; i < 1024; i += 256) {
        if (i < NCAND) { s_key[i] = -scores[(size_t)b*NCAND + i]; s_ind[i] = i; }
        else           { s_key[i] = __builtin_inff();             s_ind[i] = -1; }
    }
    __syncthreads();

    // bitonic sort, ascending by key
    for (int k = 2; k <= 1024; k <<= 1) {
        for (int j = k >> 1; j > 0; j >>= 1) {
            for (int i = tid; i < 1024; i += 256) {
                const int ixj = i ^ j;
                if (ixj > i) {
                    const bool up = ((i & k) == 0);
                    const float a = s_key[i], c = s_key[ixj];
                    if (up ? (a > c) : (a < c)) {
                        s_key[i] = c; s_key[ixj] = a;
                        const int ti = s_ind[i]; s_ind[i] = s_ind[ixj]; s_ind[ixj] = ti;
                    }
                }
            }
            __syncthreads();
        }
    }

    int count = 0;
    for (int tt = 0; tt < NCAND && count < MSLOTS; ++tt) {
        const int t = s_ind[tt];
        const float* hrow = hidden + ((size_t)b*TT + t) * HID;
        s_cvec[tid]       = hrow[tid];
        s_cvec[tid + 256] = hrow[tid + 256];
        __syncthreads();

        // L2 norm of candidate
        s_red[tid] = s_cvec[tid]*s_cvec[tid] + s_cvec[tid+256]*s_cvec[tid+256];
        __syncthreads();
        for (int st = 128; st > 0; st >>= 1) {
            if (tid < st) s_red[tid] += s_red[tid + st];
            __syncthreads();
        }
        if (tid == 0) s_bc[0] = 1.f / fmaxf(sqrtf(s_red[0]), 1e-12f);
        __syncthreads();

        // cosine sims vs selected slots (4 threads per slot)
        {
            const int m = tid >> 2, q = tid & 3;
            const float* sn = s_norm + (size_t)m * HID + q * 128;
            const float* cv = s_cvec + q * 128;
            float pp = 0.f;
            for (int d = 0; d < 128; ++d) pp += sn[d] * cv[d];
            s_red[tid] = pp;
        }
        __syncthreads();
        if (tid < MSLOTS)
            s_sims[tid] = (s_red[4*tid] + s_red[4*tid+1] + s_red[4*tid+2] + s_red[4*tid+3]) * s_bc[0];
        __syncthreads();
        if (tid == 0) {
            float mx = -__builtin_inff();
            for (int m = 0; m < count; ++m) mx = fmaxf(mx, s_sims[m]);
            const bool ok = (count == 0) || (mx <= THRESH);
            s_bc[1] = ok ? 1.f : 0.f;
        }
        __syncthreads();
        if (s_bc[1] > 0.5f) {
            const float inv = s_bc[0];
            s_norm[(size_t)count*HID + tid]       = s_cvec[tid] * inv;
            s_norm[(size_t)count*HID + tid + 256] = s_cvec[tid+256] * inv;
            float* mrow = mem + ((size_t)b*MSLOTS + count) * HID;
            mrow[tid]       = s_cvec[tid];
            mrow[tid + 256] = s_cvec[tid + 256];
            if (tid == 0) mask[b*MSLOTS + count] = 1.f;
            ++count;
        }
        __syncthreads();
    }
}

// =====================================================================
// K4: query projection + masked softmax attention over 64 slots -> ctx
// =====================================================================
__global__ void __launch_bounds__(256)
attn_kernel(const float* __restrict__ hidden, const float* __restrict__ q_w,
            const float* __restrict__ q_b, const float* __restrict__ mem,
            const float* __restrict__ mask, float* __restrict__ ctx)
{
    __shared__ float s_q[HID];
    __shared__ float s_red[256];
    __shared__ float s_att[MSLOTS];
    const int tid = threadIdx.x;
    const int b   = blockIdx.x;

    const float* qh = hidden + ((size_t)b*TT + (TT - 2)) * HID;
    for (int h = tid; h < HID; h += 256) {
        const float* wr = q_w + (size_t)h * HID;
        float a = 0.f;
        for (int k = 0; k < HID; ++k) a += qh[k] * wr[k];
        s_q[h] = a + q_b[h];
    }
    __syncthreads();

    {
        const int m = tid >> 2, q4 = tid & 3;
        const float* mr = mem + ((size_t)b*MSLOTS + m) * HID + q4 * 128;
        const float* qv = s_q + q4 * 128;
        float p = 0.f;
        for (int d = 0; d < 128; ++d) p += mr[d] * qv[d];
        s_red[tid] = p;
    }
    __syncthreads();
    if (tid < MSLOTS) {
        float s = s_red[4*tid] + s_red[4*tid+1] + s_red[4*tid+2] + s_red[4*tid+3];
        if (mask[b*MSLOTS + tid] == 0.f) s = -1e9f;
        s_att[tid] = s;
    }
    __syncthreads();
    if (tid == 0) {
        float mx = -__builtin_inff();
        for (int m = 0; m < MSLOTS; ++m) mx = fmaxf(mx, s_att[m]);
        float sum = 0.f;
        for (int m = 0; m < MSLOTS; ++m) { float e = __expf(s_att[m] - mx); s_att[m] = e; sum += e; }
        const float inv = 1.f / sum;
        for (int m = 0; m < MSLOTS; ++m) s_att[m] *= inv;
    }
    __syncthreads();
    for (int h = tid; h < HID; h += 256) {
        float a = 0.f;
        for (int m = 0; m < MSLOTS; ++m)
            a += s_att[m] * mem[((size_t)b*MSLOTS + m) * HID + h];
        ctx[(size_t)b*HID + h] = a;
    }
}

// =====================================================================
// K5: output projection  out[32,32000] = ctx[32,512] * out_w^T + out_b
// WMMA f32 16x16x4, both 16-row m-tiles per wave -> out_w read ONCE
// (64 MB single pass ~= 2.8us at 23.3 TB/s; this kernel is BW-bound)
// =====================================================================
__global__ void __launch_bounds__(256)
out_kernel(const float* __restrict__ ctx, const float* __restrict__ out_w,
           const float* __restrict__ out_b, float* __restrict__ out)
{
    const int tid  = threadIdx.x;
    const int wave = tid >> 5, lane = tid & 31;
    const int n  = lane & 15;
    const int kh = lane >> 4;
    const int n0 = (blockIdx.x * 8 + wave) * 16;

    const float* Ap0 = ctx  + (size_t)n * HID + 2*kh;          // rows 0..15
    const float* Ap1 = ctx  + (size_t)(16 + n) * HID + 2*kh;   // rows 16..31
    const float* Bp  = out_w + (size_t)(n0 + n) * HID + 2*kh;
    __builtin_prefetch(out_w + (size_t)(n0 + 16 + n) * HID, 0, 1);

    v8f acc0 = {0.f,0.f,0.f,0.f,0.f,0.f,0.f,0.f};
    v8f acc1 = {0.f,0.f,0.f,0.f,0.f,0.f,0.f,0.f};
    for (int k0 = 0; k0 < HID; k0 += 4) {
        v2f b  = *(const v2f*)(Bp  + k0);
        v2f a0 = *(const v2f*)(Ap0 + k0);
        v2f a1 = *(const v2f*)(Ap1 + k0);
        acc0 = __builtin_amdgcn_wmma_f32_16x16x4_f32(false, a0, false, b,
                                                     (short)0, acc0, false, false);
        acc1 = __builtin_amdgcn_wmma_f32_16x16x4_f32(false, a1, false, b,
                                                     (short)0, acc1, false, false);
    }
    const float bias = out_b[n0 + n];
    #pragma unroll
    for (int r = 0; r < 8; ++r) {
        out[(size_t)(r + 8*kh) * VOCAB + (n0 + n)]      = acc0[r] + bias;
        out[(size_t)(16 + r + 8*kh) * VOCAB + (n0 + n)] = acc1[r] + bias;
    }
}

// =====================================================================
extern "C" void kernel_launch(void* const* d_in, const int* in_sizes, int n_in,
                              void* d_out, int out_size, void* d_ws, size_t ws_size,
                              hipStream_t stream)
{
    (void)in_sizes; (void)n_in; (void)out_size; (void)ws_size;
    const int*   seq   = (const int*)  d_in[0];
    const float* embed = (const float*)d_in[1];
    const float* w1    = (const float*)d_in[2];
    const float* b1    = (const float*)d_in[3];
    const float* w2    = (const float*)d_in[4];
    const float* b2    = (const float*)d_in[5];
    const float* ln_g  = (const float*)d_in[6];
    const float* ln_b  = (const float*)d_in[7];
    const float* wg_w  = (const float*)d_in[8];
    const float* wg_b  = (const float*)d_in[9];
    const float* q_w   = (const float*)d_in[10];
    const float* q_b   = (const float*)d_in[11];
    const float* out_w = (const float*)d_in[12];
    const float* out_b = (const float*)d_in[13];

    float* ws     = (float*)d_ws;
    float* hidden = ws + OFF_HIDDEN;
    float* scores = ws + OFF_SCORES;
    float* mem    = ws + OFF_MEM;
    float* mask   = ws + OFF_MASK;
    float* ctx    = ws + OFF_CTX;
    float* out    = (float*)d_out;

    const size_t smem1 = (size_t)(32*HID + 32*FFD + 32*HID) * sizeof(float); // 256 KB
    hipLaunchKernelGGL(ffn_ln_kernel, dim3(BB*TT/32), dim3(256), smem1, stream,
                       seq, embed, w1, b1, w2, b2, ln_g, ln_b, hidden);

    hipLaunchKernelGGL(scores_kernel, dim3((BB*NCAND + 7)/8), dim3(256), 0, stream,
                       hidden, wg_w, wg_b, scores);

    const size_t smem2 = (size_t)(2944 + MSLOTS*HID) * sizeof(float); // ~140 KB
    hipLaunchKernelGGL(select_kernel, dim3(BB), dim3(256), smem2, stream,
                       hidden, scores, mem, mask);

    hipLaunchKernelGGL(attn_kernel, dim3(BB), dim3(256), 0, stream,
                       hidden, q_w, q_b, mem, mask, ctx);

    hipLaunchKernelGGL(out_kernel, dim3(VOCAB/16/8), dim3(256), 0, stream,
                       ctx, out_w, out_b, out);
}